// GATE_29411936043366
// MI455X (gfx1250) — compile-verified
//
#include <hip/hip_runtime.h>
#include <math.h>

typedef __attribute__((ext_vector_type(2))) float v2f;
typedef __attribute__((ext_vector_type(8))) float v8f;

#define NN 50000
#define EE 600000
#define DD 128
#define LDS_PITCH 132   // 128 + 4 pad -> conflict-free column reads (stride mod 64 banks = 4)

// ---------------------------------------------------------------------------
// zero fill
// ---------------------------------------------------------------------------
__global__ void fzero_k(float* __restrict__ p, size_t n) {
    size_t i = (size_t)blockIdx.x * blockDim.x + threadIdx.x;
    size_t stride = (size_t)gridDim.x * blockDim.x;
    for (; i < n; i += stride) p[i] = 0.0f;
}

// ---------------------------------------------------------------------------
// dropout: out = X * (mask > 0.2) / 0.8
// ---------------------------------------------------------------------------
__global__ void dropout_k(const float* __restrict__ X, const float* __restrict__ M,
                          float* __restrict__ out, size_t n) {
    size_t i = (size_t)blockIdx.x * blockDim.x + threadIdx.x;
    size_t stride = (size_t)gridDim.x * blockDim.x;
    for (; i < n; i += stride)
        out[i] = (M[i] > 0.2f) ? X[i] * 1.25f : 0.0f;
}

// ---------------------------------------------------------------------------
// degree counting + rsqrt(max(deg,1))
// ---------------------------------------------------------------------------
__global__ void deg_count_k(const int* __restrict__ src, const int* __restrict__ dst,
                            float* __restrict__ dS, float* __restrict__ dD, int e_cnt) {
    int e = blockIdx.x * blockDim.x + threadIdx.x;
    if (e >= e_cnt) return;
    atomicAdd(&dS[src[e]], 1.0f);
    atomicAdd(&dD[dst[e]], 1.0f);
}

__global__ void deg_norm_k(float* __restrict__ d, int n) {
    int i = blockIdx.x * blockDim.x + threadIdx.x;
    if (i < n) d[i] = rsqrtf(fmaxf(d[i], 1.0f));
}

// ---------------------------------------------------------------------------
// WMMA f32 GEMM:  C[nrows x 128] = f(A * rowscale) @ W (+ bias)
//   rowscale == nullptr -> no scaling;  eluIn != 0 -> apply ELU to A elements
//   bias == nullptr -> no bias.
// One 256-thread block handles a 16-row stripe; 8 waves x 16-col tiles.
// Uses V_WMMA_F32_16X16X4_F32 with K stepped 4 at a time (32 steps over K=128).
// ---------------------------------------------------------------------------
__global__ __launch_bounds__(256) void gemm_wmma_k(
        const float* __restrict__ A, const float* __restrict__ W,
        const float* __restrict__ rowscale, const float* __restrict__ bias,
        float* __restrict__ C, int eluIn) {
    __shared__ float As[16 * LDS_PITCH];

    const int tid = threadIdx.x;
    const int rowBase = blockIdx.x * 16;

    // stage A tile (16 x 128) into LDS with fused scale / ELU
#pragma unroll
    for (int i = 0; i < 8; ++i) {
        int idx = tid + i * 256;            // 0 .. 2047
        int r = idx >> 7;
        int c = idx & 127;
        float v = A[(size_t)(rowBase + r) * DD + c];
        if (rowscale) v *= rowscale[rowBase + r];
        if (eluIn)    v = (v > 0.0f) ? v : (expf(v) - 1.0f);
        As[r * LDS_PITCH + c] = v;
    }
    __syncthreads();

    const int wave = tid >> 5;          // 0..7 -> column tile
    const int lane = tid & 31;
    const int hl   = lane & 15;         // half-lane index
    const int kh   = lane >> 4;         // 0 or 1 (K pair select)
    const int nCol = wave * 16 + hl;    // output column (0..127)

    v8f acc = {};                        // zero accumulator
#pragma unroll 4
    for (int kb = 0; kb < DD; kb += 4) {
        const int k0 = kb + kh * 2;      // lanes 0-15: K=kb,kb+1 | lanes 16-31: K=kb+2,kb+3
        v2f a, b;
        a.x = As[hl * LDS_PITCH + k0];       // A[M=hl][k0]
        a.y = As[hl * LDS_PITCH + k0 + 1];   // A[M=hl][k0+1]
        b.x = W[(size_t)k0 * DD + nCol];     // W[k0][nCol]
        b.y = W[(size_t)(k0 + 1) * DD + nCol];
        acc = __builtin_amdgcn_wmma_f32_16x16x4_f32(
                /*neg_a=*/false, a, /*neg_b=*/false, b,
                /*c_mod=*/(short)0, acc, /*reuse_a=*/false, /*reuse_b=*/false);
    }

    const float bv = bias ? bias[nCol] : 0.0f;
#pragma unroll
    for (int r = 0; r < 8; ++r) {
        const int m = r + kh * 8;        // acc VGPR r -> rows r (lanes 0-15) / r+8 (lanes 16-31)
        C[(size_t)(rowBase + m) * DD + nCol] = acc[r] + bv;
    }
}

// ---------------------------------------------------------------------------
// edge scatter:  agg[dst[e]] += h[src[e]]   (one wave32 per edge, float4/lane)
// ---------------------------------------------------------------------------
__global__ void edge_scatter_k(const float* __restrict__ h,
                               const int* __restrict__ src, const int* __restrict__ dst,
                               float* __restrict__ agg, int e_cnt) {
    size_t g = (size_t)blockIdx.x * blockDim.x + threadIdx.x;
    int e = (int)(g >> 5);
    if (e >= e_cnt) return;
    int lane = (int)(g & 31);
    const float4 v = ((const float4*)(h + (size_t)src[e] * DD))[lane];
    float* a = agg + (size_t)dst[e] * DD + lane * 4;
    atomicAdd(a + 0, v.x);
    atomicAdd(a + 1, v.y);
    atomicAdd(a + 2, v.z);
    atomicAdd(a + 3, v.w);
}

// ---------------------------------------------------------------------------
// post-aggregation:  x = f(agg * norm_dst + b), f = ELU or identity
// ---------------------------------------------------------------------------
__global__ void post_layer_k(const float* __restrict__ agg, const float* __restrict__ nd,
                             const float* __restrict__ b, float* __restrict__ x,
                             int doElu, size_t n) {
    size_t i = (size_t)blockIdx.x * blockDim.x + threadIdx.x;
    size_t stride = (size_t)gridDim.x * blockDim.x;
    for (; i < n; i += stride) {
        size_t r = i >> 7;
        int    c = (int)(i & 127);
        float v = agg[i] * nd[r] + b[c];
        if (doElu) v = (v > 0.0f) ? v : (expf(v) - 1.0f);
        x[i] = v;
    }
}

// ---------------------------------------------------------------------------
// column stats: stats[0..127]=sum, stats[128..255]=sumsq
// ---------------------------------------------------------------------------
__global__ __launch_bounds__(256) void col_partial_k(const float* __restrict__ emb,
                                                     float* __restrict__ stats, int nrows) {
    __shared__ float sh[512];
    const int tid  = threadIdx.x;
    const int c    = tid & 127;
    const int half = tid >> 7;
    float s = 0.0f, s2 = 0.0f;
    for (int r = blockIdx.x * 2 + half; r < nrows; r += gridDim.x * 2) {
        float v = emb[(size_t)r * DD + c];
        s += v; s2 += v * v;
    }
    sh[tid] = s;
    sh[256 + tid] = s2;
    __syncthreads();
    if (tid < 128) {
        atomicAdd(&stats[c],       sh[tid]       + sh[tid + 128]);
        atomicAdd(&stats[128 + c], sh[256 + tid] + sh[256 + tid + 128]);
    }
}

// stats[256..383]=mean, stats[384..511]=1/std (unbiased, ddof=1)
__global__ void col_final_k(float* __restrict__ stats, int nrows) {
    int c = threadIdx.x;
    if (c >= 128) return;
    float sum  = stats[c];
    float sq   = stats[128 + c];
    float mean = sum / (float)nrows;
    float var  = (sq - sum * mean) / (float)(nrows - 1);
    stats[256 + c] = mean;
    stats[384 + c] = rsqrtf(var);
}

__global__ void col_norm_k(float* __restrict__ emb, const float* __restrict__ stats, size_t n) {
    size_t i = (size_t)blockIdx.x * blockDim.x + threadIdx.x;
    size_t stride = (size_t)gridDim.x * blockDim.x;
    for (; i < n; i += stride) {
        int c = (int)(i & 127);
        emb[i] = (emb[i] - stats[256 + c]) * stats[384 + c];
    }
}

// ---------------------------------------------------------------------------
// classifier: out[r, 0:2] = emb[r, :] @ Wc[128x2] + bc
// ---------------------------------------------------------------------------
__global__ void cls2_k(const float* __restrict__ emb, const float* __restrict__ Wc,
                       const float* __restrict__ bc, float* __restrict__ out, int nrows) {
    int r = blockIdx.x * blockDim.x + threadIdx.x;
    if (r >= nrows) return;
    const float* e = emb + (size_t)r * DD;
    float a0 = 0.0f, a1 = 0.0f;
#pragma unroll 4
    for (int c = 0; c < DD; ++c) {
        float v = e[c];
        a0 += v * Wc[c * 2 + 0];
        a1 += v * Wc[c * 2 + 1];
    }
    out[(size_t)r * 2 + 0] = a0 + bc[0];
    out[(size_t)r * 2 + 1] = a1 + bc[1];
}

// ---------------------------------------------------------------------------
// host-side orchestration
// ---------------------------------------------------------------------------
static inline void zero_f(float* p, size_t n, hipStream_t s) {
    int blocks = (int)((n + 255 * 4) / (256 * 4));
    if (blocks < 1) blocks = 1;
    if (blocks > 4096) blocks = 4096;
    fzero_k<<<blocks, 256, 0, s>>>(p, n);
}

extern "C" void kernel_launch(void* const* d_in, const int* in_sizes, int n_in,
                              void* d_out, int out_size, void* d_ws, size_t ws_size,
                              hipStream_t stream) {
    (void)in_sizes; (void)n_in; (void)out_size; (void)ws_size;

    const float* X[2]    = { (const float*)d_in[0], (const float*)d_in[1] };
    const float* MSK[2]  = { (const float*)d_in[2], (const float*)d_in[3] };
    const int*   SRC[2]  = { (const int*)d_in[4],  (const int*)d_in[6]  };
    const int*   DST[2]  = { (const int*)d_in[5],  (const int*)d_in[7]  };
    const float* Wg[3]   = { (const float*)d_in[8], (const float*)d_in[10], (const float*)d_in[12] };
    const float* Bg[3]   = { (const float*)d_in[9], (const float*)d_in[11], (const float*)d_in[13] };
    const float* Wl      = (const float*)d_in[14];
    const float* bl      = (const float*)d_in[15];
    const float* Wc      = (const float*)d_in[16];
    const float* bc      = (const float*)d_in[17];

    float* out = (float*)d_out;
    float* emb_out[2] = { out,                    out + (size_t)NN * DD };
    float* cls_out[2] = { out + 2ull * NN * DD,   out + 2ull * NN * DD + (size_t)NN * 2 };

    // workspace layout (floats)
    float* ws    = (float*)d_ws;
    float* buf0  = ws;                              // x        [N*128]
    float* buf1  = ws +  (size_t)NN * DD;           // h        [N*128]
    float* buf2  = ws + 2ull * NN * DD;             // agg      [N*128]
    float* ns    = ws + 3ull * NN * DD;             // norm_src [N]
    float* nd    = ns + NN;                         // norm_dst [N]
    float* stats = nd + NN;                         // [512]

    const size_t nelem = (size_t)NN * DD;
    const int gemm_grid    = NN / 16;                       // 3125 (exact)
    const int scatter_grid = (int)(((size_t)EE * 32 + 255) / 256);
    const int elem_grid    = 4096;

    for (int br = 0; br < 2; ++br) {
        // --- degree norms ---
        zero_f(ns, NN, stream);
        zero_f(nd, NN, stream);
        deg_count_k<<<(EE + 255) / 256, 256, 0, stream>>>(SRC[br], DST[br], ns, nd, EE);
        deg_norm_k<<<(NN + 255) / 256, 256, 0, stream>>>(ns, NN);
        deg_norm_k<<<(NN + 255) / 256, 256, 0, stream>>>(nd, NN);

        // --- dropout ---
        dropout_k<<<elem_grid, 256, 0, stream>>>(X[br], MSK[br], buf0, nelem);

        // --- 3 GraphConv layers ---
        for (int L = 0; L < 3; ++L) {
            gemm_wmma_k<<<gemm_grid, 256, 0, stream>>>(buf0, Wg[L], ns, nullptr, buf1, 0);
            zero_f(buf2, nelem, stream);
            edge_scatter_k<<<scatter_grid, 256, 0, stream>>>(buf1, SRC[br], DST[br], buf2, EE);
            post_layer_k<<<elem_grid, 256, 0, stream>>>(buf2, nd, Bg[L], buf0, (L < 2) ? 1 : 0, nelem);
        }

        // --- emb = ELU(x3) @ Wl + bl  (ELU fused into A load) ---
        gemm_wmma_k<<<gemm_grid, 256, 0, stream>>>(buf0, Wl, nullptr, bl, emb_out[br], 1);

        // --- column standardization (ddof=1) ---
        zero_f(stats, 512, stream);
        col_partial_k<<<512, 256, 0, stream>>>(emb_out[br], stats, NN);
        col_final_k<<<1, 128, 0, stream>>>(stats, NN);
        col_norm_k<<<elem_grid, 256, 0, stream>>>(emb_out[br], stats, nelem);

        // --- classifier head ---
        cls2_k<<<(NN + 255) / 256, 256, 0, stream>>>(emb_out[br], Wc, bc, cls_out[br], NN);
    }
}